// VectorMapNet_506806141485
// MI455X (gfx1250) — compile-verified
//
#include <hip/hip_runtime.h>
#include <stdint.h>

// ---------------- problem constants ----------------
#define BATCH  32
#define NCH    65
#define HC     50
#define WC     100
#define NCELL  5000        // HC*WC
#define SCELL  8
#define KTOP   300
#define HPIX   400
#define WPIX   800
#define BORDER 8
#define PLANE  (HPIX * WPIX)   // 320000

// output layout (floats): pos[32,300,3] | dt[32,300,64] | mask[32,300]
#define POS_BASE  0
#define DT_BASE   (BATCH * KTOP * 3)              // 28800
#define MASK_BASE (DT_BASE + BATCH * KTOP * 64)   // 643200

#define BDIMB 1024
#define SORTN 4096
#define CHUNK 2500
#define DTN   (KTOP * 64)                          // 19200
#define NITER ((DTN + BDIMB - 1) / BDIMB)          // 19

// order-preserving float<->uint mapping (monotonic, invertible)
__device__ __forceinline__ uint32_t f32_ord(float f) {
  uint32_t u = __float_as_uint(f);
  return (u & 0x80000000u) ? ~u : (u | 0x80000000u);
}
__device__ __forceinline__ float ord_f32(uint32_t o) {
  uint32_t u = (o & 0x80000000u) ? (o & 0x7FFFFFFFu) : ~o;
  return __uint_as_float(u);
}

// ---------------------------------------------------------------------------
// Kernel A: per-cell online softmax (single pass over 65 channels), argmax,
// score, coords, validity. One thread per cell; consecutive lanes read
// consecutive cells -> fully coalesced 128B lines per channel step.
// ---------------------------------------------------------------------------
__global__ void __launch_bounds__(256) cell_kernel(
    const float* __restrict__ vertex,
    float* __restrict__ sc_ws, int* __restrict__ meta_ws) {
  int cell = blockIdx.x * blockDim.x + threadIdx.x;
  if (cell >= BATCH * NCELL) return;
  int b = cell / NCELL;
  int i = cell - b * NCELL;
  const float* base = vertex + (size_t)b * NCH * NCELL + i;

  float m = -__builtin_inff();   // running max
  float ssum = 0.f;              // running sum of exp(x - m)
  float x63 = 0.f;               // raw value of channel 63 (needed if dust wins)
  int   am = 0;                  // first-occurrence argmax
  for (int c = 0; c < NCH; ++c) {
    float x = base[(size_t)c * NCELL];
    if (c == 63) x63 = x;
    if (x > m) { ssum *= __expf(m - x); m = x; am = c; }  // first iter: 0*exp(-inf)=0
    ssum += __expf(x - m);
  }
  bool  vc     = (am < 64);                 // not the dust channel
  int   ch     = vc ? am : 63;              // min(amax, 63)
  float chosen = vc ? m : x63;
  float score  = __expf(chosen - m) / ssum; // softmax value of chosen channel

  int hc  = i / WC, wc = i - hc * WC;
  int row = hc * SCELL + (ch >> 3);
  int col = wc * SCELL + (ch & 7);
  bool bok = (row >= BORDER) && (row < HPIX - BORDER) &&
             (col >= BORDER) && (col < WPIX - BORDER);
  bool v = vc && bok;

  sc_ws[cell]   = score;
  meta_ws[cell] = col | (row << 10) | (vc ? (1 << 20) : 0) | (v ? (1 << 21) : 0);
}

// ---------------------------------------------------------------------------
// Kernel B: one block per batch. Stable-rank scan, exact top-300 selection
// (chunked bitonic sort + bitonic merge on 64-bit keys), outputs pos/mask,
// then data-dependent distance-patch gather via double-buffered CDNA5 async
// global->LDS copies (ASYNCcnt pipelining) for the dt embedding.
// ---------------------------------------------------------------------------
__global__ void __launch_bounds__(BDIMB) select_kernel(
    const float* __restrict__ distance,
    const float* __restrict__ sc_ws, const int* __restrict__ meta_ws,
    float* __restrict__ out) {
  __shared__ unsigned long long keys[SORTN];   // 32 KB; reused as async staging
  __shared__ unsigned long long side[1024];    //  8 KB
  __shared__ int cntV[BDIMB];                  //  4 KB
  __shared__ int cntC[BDIMB];                  //  4 KB
  __shared__ int orderV[KTOP];
  __shared__ int orderC[KTOP];
  __shared__ int nVs, nCs;
  // total static LDS ~50.6 KB

  const int tid = threadIdx.x;
  const int b   = blockIdx.x;
  const float* sc   = sc_ws  + b * NCELL;
  const int*   meta = meta_ws + b * NCELL;

  // ---- stable ranks: 5 contiguous cells per thread + block inclusive scan ----
  int localV = 0, localC = 0;
  const int c0 = tid * 5;
  #pragma unroll
  for (int u = 0; u < 5; ++u) {
    int ci = c0 + u;
    if (ci < NCELL) {
      int mt = meta[ci];
      localC += (mt >> 20) & 1;
      localV += (mt >> 21) & 1;
    }
  }
  cntV[tid] = localV; cntC[tid] = localC;
  __syncthreads();
  for (int off = 1; off < BDIMB; off <<= 1) {   // Hillis-Steele inclusive scan
    int av = (tid >= off) ? cntV[tid - off] : 0;
    int ac = (tid >= off) ? cntC[tid - off] : 0;
    __syncthreads();
    cntV[tid] += av; cntC[tid] += ac;
    __syncthreads();
  }
  int baseV = cntV[tid] - localV;
  int baseC = cntC[tid] - localC;
  if (tid == BDIMB - 1) { nVs = cntV[tid]; nCs = cntC[tid]; }
  #pragma unroll
  for (int u = 0; u < 5; ++u) {
    int ci = c0 + u;
    if (ci < NCELL) {
      int mt = meta[ci];
      if ((mt >> 20) & 1) { if (baseC < KTOP) orderC[baseC] = ci; ++baseC; }
      if ((mt >> 21) & 1) { if (baseV < KTOP) orderV[baseV] = ci; ++baseV; }
    }
  }
  __syncthreads();
  const int nV = nVs, nC = nCs;

  // ---- exact top-300: sort two 2500-cell chunks (desc), keep top-512 each ----
  // key = ord(score)<<32 | ~cellIdx  (desc score, asc index tie-break); 0 = invalid
  for (int chunk = 0; chunk < 2; ++chunk) {
    for (int idx = tid; idx < SORTN; idx += BDIMB) {
      unsigned long long key = 0ull;
      int ci = chunk * CHUNK + idx;
      if (idx < CHUNK && ci < NCELL) {
        int mt = meta[ci];
        if ((mt >> 21) & 1)
          key = ((unsigned long long)f32_ord(sc[ci]) << 32) | (uint32_t)(~ci);
      }
      keys[idx] = key;
    }
    for (int k = 2; k <= SORTN; k <<= 1) {        // bitonic sort, descending
      for (int j = k >> 1; j > 0; j >>= 1) {
        __syncthreads();
        for (int idx = tid; idx < SORTN; idx += BDIMB) {
          int partner = idx ^ j;
          if (partner > idx) {
            bool desc = ((idx & k) == 0);
            unsigned long long a = keys[idx], bb = keys[partner];
            if ((a < bb) == desc) { keys[idx] = bb; keys[partner] = a; }
          }
        }
      }
    }
    __syncthreads();
    if (tid < 512) side[chunk * 512 + tid] = keys[tid];
    __syncthreads();
  }
  // merge: [desc 512 | asc 512] is bitonic -> full descending merge of 1024
  {
    unsigned long long vkey = (tid < 512) ? side[tid] : side[1023 - (tid - 512)];
    keys[tid] = vkey;
    __syncthreads();
    for (int j = 512; j > 0; j >>= 1) {
      if ((tid & j) == 0) {
        unsigned long long a = keys[tid], bb = keys[tid + j];
        if (a < bb) { keys[tid] = bb; keys[tid + j] = a; }
      }
      __syncthreads();
    }
  }

  // ---- pos + mask ----
  const bool use_topk = (nV > KTOP);
  if (tid < KTOP) {
    float score = 0.f, msk = 0.f;
    int x = 0, y = 0;
    if (use_topk) {
      unsigned long long key = keys[tid];
      int ci = (int)(~(uint32_t)key);
      score = ord_f32((uint32_t)(key >> 32));    // exact score bits
      int mt = meta[ci];
      x = mt & 1023; y = (mt >> 10) & 511;
      msk = 1.f;
    } else if (tid < nV) {
      int ci = orderV[tid];
      score = sc[ci];
      int mt = meta[ci];
      x = mt & 1023; y = (mt >> 10) & 511;
      msk = 1.f;
    }
    float* pos = out + POS_BASE + ((size_t)b * KTOP + tid) * 3;
    pos[0] = ((float)x - 400.f) / 400.f;   // (x - W/2)/(W/2); padded -> -1
    pos[1] = ((float)y - 200.f) / 200.f;   // (y - H/2)/(H/2); padded -> -1
    pos[2] = score;
    out[MASK_BASE + b * KTOP + tid] = msk;
  }
  __syncthreads();   // keys region is dead past this point -> reuse as staging

  // ---- dt gather: 300 cells x 64 px x 3 ch, double-buffered async->LDS ----
  const int nCeff = (nC < KTOP) ? nC : KTOP;
  const float* dist_b = distance + (size_t)b * 3 * PLANE;
  float* stagef = (float*)keys;   // 2 buffers x 1024 lanes x 3 floats = 24 KB

  // issue one element's 3 channel loads into staging buffer `buf`
  auto issue = [&](int t0, int buf) {
    int t = t0 + tid;
    int k = t >> 6, j = t & 63;
    bool valid = (t < DTN) && (k < nCeff);
    size_t pix = 0;
    if (valid) {
      int ci = orderC[k];
      int hc = ci / WC, wc = ci - hc * WC;
      int r = hc * SCELL + (j >> 3);
      int c = wc * SCELL + (j & 7);
      pix = (size_t)r * WPIX + c;
    }
    unsigned long long g0 = (unsigned long long)(uintptr_t)(dist_b + pix);
    unsigned long long g1 = g0 + (unsigned long long)PLANE * 4u;
    unsigned long long g2 = g1 + (unsigned long long)PLANE * 4u;
    uint32_t lds0 = (uint32_t)(uintptr_t)&stagef[(buf * BDIMB + tid) * 3];
    asm volatile("global_load_async_to_lds_b32 %0, %1, off"
                 :: "v"(lds0),      "v"(g0) : "memory");
    asm volatile("global_load_async_to_lds_b32 %0, %1, off"
                 :: "v"(lds0 + 4u), "v"(g1) : "memory");
    asm volatile("global_load_async_to_lds_b32 %0, %1, off"
                 :: "v"(lds0 + 8u), "v"(g2) : "memory");
  };

  issue(0, 0);                                   // prologue
  for (int it = 0; it < NITER; ++it) {
    if (it + 1 < NITER) {
      issue((it + 1) * BDIMB, (it + 1) & 1);     // keep next tile in flight
      // async loads complete in order: <=3 outstanding means tile `it` landed
      asm volatile("s_wait_asynccnt 0x3" ::: "memory");
    } else {
      asm volatile("s_wait_asynccnt 0x0" ::: "memory");
    }
    int t = it * BDIMB + tid;
    int k = t >> 6;
    if (t < DTN) {
      float val = 0.f;
      if (k < nCeff) {
        int s0 = ((it & 1) * BDIMB + tid) * 3;
        float a = stagef[s0 + 0];
        float bch = stagef[s0 + 1];
        float cch = stagef[s0 + 2];
        val = fmaxf(a, fmaxf(bch, cch)) * 0.1f;  // max over channels / THR
      }
      out[DT_BASE + (size_t)b * DTN + t] = val;
    }
  }
}

// ---------------------------------------------------------------------------
extern "C" void kernel_launch(void* const* d_in, const int* in_sizes, int n_in,
                              void* d_out, int out_size, void* d_ws, size_t ws_size,
                              hipStream_t stream) {
  const float* vertex   = (const float*)d_in[0];   // [32,65,50,100] f32
  const float* distance = (const float*)d_in[1];   // [32,3,400,800] f32
  float* out = (float*)d_out;

  float* sc_ws  = (float*)d_ws;                                  // 640 KB
  int*   meta_ws = (int*)((char*)d_ws + (size_t)BATCH * NCELL * sizeof(float)); // 640 KB

  int cells = BATCH * NCELL;
  cell_kernel<<<(cells + 255) / 256, 256, 0, stream>>>(vertex, sc_ws, meta_ws);
  select_kernel<<<BATCH, BDIMB, 0, stream>>>(distance, sc_ws, meta_ws, out);
}